// Sampler_5239860101341
// MI455X (gfx1250) — compile-verified
//
#include <hip/hip_runtime.h>
#include <hip/hip_bf16.h>
#include <math.h>

#define BROWS 128
#define VOCAB 128000
#define N4 (VOCAB / 4)
#define NT 1024
#define NW (NT / 32)
#define SAMP_EPS 1e-5f
#define NEG_INF (-__builtin_inff())

// d_out layout (float*, int outputs stored as raw int bits):
//   [0,128)        sampled (int32)
//   [128,256)      token_lp (f32)
//   [256,1280)     topk_vals (f32) [128][8]
//   [1280,2304)    topk_idx (int32) [128][8]
//   [2304, ...)    final_logprobs (f32) [128][128000]
#define OUT_TLP  (BROWS)
#define OUT_T8V  (2 * BROWS)
#define OUT_T8I  (2 * BROWS + 8 * BROWS)
#define OUT_FLP  (2 * BROWS + 16 * BROWS)

#if defined(__HIP_DEVICE_COMPILE__) &&                                         \
    __has_builtin(__builtin_amdgcn_global_load_async_to_lds_b128) &&           \
    __has_builtin(__builtin_amdgcn_s_wait_asynccnt)
#define USE_ASYNC_LDS 1
typedef int v4i_t __attribute__((ext_vector_type(4)));
typedef __attribute__((address_space(1))) v4i_t* gp4_t;  // global int4*
typedef __attribute__((address_space(3))) v4i_t* lp4_t;  // LDS int4*
#else
#define USE_ASYNC_LDS 0
#endif

// Monotonic float -> uint key (order-preserving for all finite floats)
__device__ __forceinline__ unsigned key_of(float f) {
  unsigned u = __float_as_uint(f);
  return u ^ ((unsigned)((int)u >> 31) | 0x80000000u);
}

// Merge two descending-sorted 8-lists, keep top 8 (ties -> lower index)
__device__ __forceinline__ void merge8(float av[8], int ai[8],
                                       const float bv[8], const int bi[8]) {
  float ov[8]; int oi[8];
  int pa = 0, pb = 0;
#pragma unroll
  for (int o = 0; o < 8; ++o) {
    float x = av[pa], y = bv[pb];
    bool ta = (x > y) || (x == y && ai[pa] <= bi[pb]);
    ov[o] = ta ? x : y;
    oi[o] = ta ? ai[pa] : bi[pb];
    pa += ta ? 1 : 0;
    pb += ta ? 0 : 1;
  }
#pragma unroll
  for (int o = 0; o < 8; ++o) { av[o] = ov[o]; ai[o] = oi[o]; }
}

// wave32 shuffle tree-merge of per-lane top-8 lists -> lane 0 has wave top-8
__device__ __forceinline__ void wave_merge8(float av[8], int ai[8]) {
#pragma unroll
  for (int off = 16; off > 0; off >>= 1) {
    float bv[8]; int bi[8];
#pragma unroll
    for (int j = 0; j < 8; ++j) {
      bv[j] = __shfl_down(av[j], (unsigned)off, 32);
      bi[j] = __shfl_down(ai[j], (unsigned)off, 32);
    }
    merge8(av, ai, bv, bi);
  }
}

__global__ __launch_bounds__(NT) void sampler_kernel(
    const float* __restrict__ logits, const float* __restrict__ temperature,
    const int* __restrict__ top_k, const float* __restrict__ top_p,
    const float* __restrict__ noise, float* __restrict__ out) {
  const int row  = blockIdx.x;
  const int tid  = threadIdx.x;
  const int lane = tid & 31;
  const int wid  = tid >> 5;

  __shared__ unsigned s_hist[256];   // radix histogram (also reused as float)
  __shared__ float    s_rA[NW];
  __shared__ float    s_rB[NW];
  __shared__ float    s_rD[NW];
  __shared__ int      s_rC[NW];
  __shared__ float    s_w8v[NW * 8];
  __shared__ int      s_w8i[NW * 8];
  __shared__ unsigned s_selU[2];
  __shared__ float    s_selF[2];
#if USE_ASYNC_LDS
  __shared__ float4   s_stage[2][NT];  // async noise staging, double buffered
#endif

  const size_t  base = (size_t)row * VOCAB;
  const float4* L4   = (const float4*)(logits + base);
  const float4* U4   = (const float4*)(noise + base);

  const float temp  = temperature[row];
  const float tsafe = (temp < SAMP_EPS) ? 1.0f : temp;
  int kk = top_k[row];
  kk = kk < 1 ? 1 : (kk > VOCAB ? VOCAB : kk);
  const float tp = top_p[row];

  // ---------------- Pass 1: raw logsumexp + argmax + per-lane top-8 --------
  float m = NEG_INF, s = 0.0f;
  float bestv = NEG_INF; int besti = 0x7fffffff;
  float t8v[8]; int t8i[8];
#pragma unroll
  for (int j = 0; j < 8; ++j) { t8v[j] = NEG_INF; t8i[j] = 0x7fffffff; }

  for (int i = tid; i < N4; i += NT) {
    __builtin_prefetch((const void*)(L4 + i + 4 * NT), 0, 0);
    float4 v = L4[i];
    float fv[4] = {v.x, v.y, v.z, v.w};
#pragma unroll
    for (int c = 0; c < 4; ++c) {
      float f = fv[c];
      int idx = 4 * i + c;
      if (f > m) { s = s * expf(m - f) + 1.0f; m = f; }
      else       { s += expf(f - m); }
      if (f > bestv) { bestv = f; besti = idx; }
      if (f > t8v[7]) {  // within a lane, idx only increases -> strict > OK
        int j = 7;
        while (j > 0 && f > t8v[j - 1]) {
          t8v[j] = t8v[j - 1]; t8i[j] = t8i[j - 1]; --j;
        }
        t8v[j] = f; t8i[j] = idx;
      }
    }
  }

  // wave32 reductions
#pragma unroll
  for (int off = 16; off > 0; off >>= 1) {
    float m2 = __shfl_down(m, (unsigned)off, 32);
    float s2 = __shfl_down(s, (unsigned)off, 32);
    float M  = fmaxf(m, m2);
    s = s * expf(m - M) + s2 * expf(m2 - M);
    m = M;
    float v2 = __shfl_down(bestv, (unsigned)off, 32);
    int   i2 = __shfl_down(besti, (unsigned)off, 32);
    if (v2 > bestv || (v2 == bestv && i2 < besti)) { bestv = v2; besti = i2; }
  }
  wave_merge8(t8v, t8i);

  if (lane == 0) {
    s_rA[wid] = m; s_rB[wid] = s; s_rD[wid] = bestv; s_rC[wid] = besti;
#pragma unroll
    for (int j = 0; j < 8; ++j) { s_w8v[wid * 8 + j] = t8v[j]; s_w8i[wid * 8 + j] = t8i[j]; }
  }
  __syncthreads();
  if (wid == 0) {
    m = s_rA[lane]; s = s_rB[lane]; bestv = s_rD[lane]; besti = s_rC[lane];
#pragma unroll
    for (int j = 0; j < 8; ++j) { t8v[j] = s_w8v[lane * 8 + j]; t8i[j] = s_w8i[lane * 8 + j]; }
#pragma unroll
    for (int off = 16; off > 0; off >>= 1) {
      float m2 = __shfl_down(m, (unsigned)off, 32);
      float s2 = __shfl_down(s, (unsigned)off, 32);
      float M  = fmaxf(m, m2);
      s = s * expf(m - M) + s2 * expf(m2 - M);
      m = M;
      float v2 = __shfl_down(bestv, (unsigned)off, 32);
      int   i2 = __shfl_down(besti, (unsigned)off, 32);
      if (v2 > bestv || (v2 == bestv && i2 < besti)) { bestv = v2; besti = i2; }
    }
    wave_merge8(t8v, t8i);
    if (lane == 0) {
      s_rA[0] = m; s_rB[0] = s; s_rC[0] = besti;
#pragma unroll
      for (int j = 0; j < 8; ++j) { s_w8v[j] = t8v[j]; s_w8i[j] = t8i[j]; }
    }
  }
  __syncthreads();
  const float M0  = s_rA[0];
  const float lse = M0 + logf(s_rB[0]);
  const int greedyIdx = s_rC[0];
  const float Mx = M0 / tsafe;  // division is monotone for tsafe>0 -> exact max of x
  __syncthreads();

  // ---------------- top-k: 4-pass 8-bit radix select (counts, LDS atomics) -
  unsigned prefix = 0;
  unsigned need = (unsigned)kk;
  for (int pass = 0; pass < 4; ++pass) {
    const int shift = (3 - pass) * 8;
    if (tid < 256) s_hist[tid] = 0u;
    __syncthreads();
    for (int i = tid; i < N4; i += NT) {
      float4 v = L4[i];
      float fv[4] = {v.x, v.y, v.z, v.w};
#pragma unroll
      for (int c = 0; c < 4; ++c) {
        unsigned key = key_of(fv[c] / tsafe);
        if (pass == 0 || (key >> (shift + 8)) == prefix)
          atomicAdd(&s_hist[(key >> shift) & 0xFFu], 1u);
      }
    }
    __syncthreads();
    if (tid == 0) {
      unsigned cum = 0, sel = 0;
      for (int b = 255; b >= 0; --b) {
        unsigned h = s_hist[b];
        if (need > cum && need <= cum + h) { sel = (unsigned)b; break; }
        cum += h;
      }
      s_selU[0] = sel; s_selU[1] = cum;
    }
    __syncthreads();
    prefix = (prefix << 8) | s_selU[0];
    need  -= s_selU[1];
    __syncthreads();
  }
  const unsigned TkKey = prefix;  // key of k-th largest x; keep key >= TkKey

  // -------- top-p: 4-pass weighted radix select (exp weights, ds_add_f32) --
  unsigned TpKey = 0;
  if (tp < 1.0f) {
    float* fh = (float*)s_hist;
    unsigned pprefix = 0;
    for (int pass = 0; pass < 4; ++pass) {
      const int shift = (3 - pass) * 8;
      if (tid < 256) fh[tid] = 0.0f;
      __syncthreads();
      for (int i = tid; i < N4; i += NT) {
        float4 v = L4[i];
        float fv[4] = {v.x, v.y, v.z, v.w};
#pragma unroll
        for (int c = 0; c < 4; ++c) {
          float x = fv[c] / tsafe;
          unsigned key = key_of(x);
          if (key >= TkKey && (pass == 0 || (key >> (shift + 8)) == pprefix))
            atomicAdd(&fh[(key >> shift) & 0xFFu], expf(x - Mx));
        }
      }
      __syncthreads();
      if (tid == 0) {
        float target;
        if (pass == 0) {
          float Z = 0.0f;
          for (int b = 0; b < 256; ++b) Z += fh[b];
          target = tp * Z;
        } else {
          target = s_selF[0];
        }
        float run = 0.0f; int sel = -1;
        for (int b = 255; b >= 0; --b) {
          float h = fh[b];
          if (run + h > target) { sel = b; break; }
          run += h;
        }
        if (sel < 0) { sel = 0; run = 0.0f; }  // float-sum edge: keep everything
        s_selU[0] = (unsigned)sel;
        s_selF[0] = target - run;
      }
      __syncthreads();
      pprefix = (pprefix << 8) | s_selU[0];
      __syncthreads();
    }
    TpKey = pprefix;
  }
  const unsigned TKey = (TkKey > TpKey) ? TkKey : TpKey;

  // ---- Pass: filtered partition Z2 + gumbel argmax (noise via async->LDS) --
  float z2 = 0.0f;
  float gv = NEG_INF; int gi = 0x7fffffff;
#if USE_ASYNC_LDS
  __builtin_amdgcn_global_load_async_to_lds_b128(
      (gp4_t)(U4 + tid), (lp4_t)&s_stage[0][tid], 0, 0);
#endif
  int it = 0;
  for (int i = tid; i < N4; i += NT, ++it) {
    float4 uv;
#if USE_ASYNC_LDS
    int inext = i + NT;  // 32000 % 32 == 0 -> wave-uniform branch
    if (inext < N4) {
      __builtin_amdgcn_global_load_async_to_lds_b128(
          (gp4_t)(U4 + inext), (lp4_t)&s_stage[(it + 1) & 1][tid], 0, 0);
      __builtin_amdgcn_s_wait_asynccnt(1);
    } else {
      __builtin_amdgcn_s_wait_asynccnt(0);
    }
    uv = s_stage[it & 1][tid];
#else
    uv = U4[i];
#endif
    float4 lv = L4[i];
    float fu[4] = {uv.x, uv.y, uv.z, uv.w};
    float fl[4] = {lv.x, lv.y, lv.z, lv.w};
#pragma unroll
    for (int c = 0; c < 4; ++c) {
      float x = fl[c] / tsafe;
      unsigned key = key_of(x);
      if (key >= TKey) {
        z2 += expf(x - Mx);
        float g = -logf(-logf(fu[c]));
        float cand = x + g;
        int idx = 4 * i + c;
        if (cand > gv || (cand == gv && idx < gi)) { gv = cand; gi = idx; }
      }
    }
  }
#pragma unroll
  for (int off = 16; off > 0; off >>= 1) {
    z2 += __shfl_down(z2, (unsigned)off, 32);
    float v2 = __shfl_down(gv, (unsigned)off, 32);
    int   i2 = __shfl_down(gi, (unsigned)off, 32);
    if (v2 > gv || (v2 == gv && i2 < gi)) { gv = v2; gi = i2; }
  }
  if (lane == 0) { s_rA[wid] = z2; s_rB[wid] = gv; s_rC[wid] = gi; }
  __syncthreads();
  if (wid == 0) {
    z2 = s_rA[lane]; gv = s_rB[lane]; gi = s_rC[lane];
#pragma unroll
    for (int off = 16; off > 0; off >>= 1) {
      z2 += __shfl_down(z2, (unsigned)off, 32);
      float v2 = __shfl_down(gv, (unsigned)off, 32);
      int   i2 = __shfl_down(gi, (unsigned)off, 32);
      if (v2 > gv || (v2 == gv && i2 < gi)) { gv = v2; gi = i2; }
    }
    if (lane == 0) { s_rA[0] = z2; s_rC[0] = gi; }
  }
  __syncthreads();
  const float lZ2 = logf(s_rA[0]);
  const int randIdx = s_rC[0];

  // ---------------- Pass: write final_logprobs (b128 stores) ---------------
  float4* O4 = (float4*)(out + OUT_FLP + base);
  for (int i = tid; i < N4; i += NT) {
    float4 lv = L4[i];
    float fl[4] = {lv.x, lv.y, lv.z, lv.w};
    float ov[4];
#pragma unroll
    for (int c = 0; c < 4; ++c) {
      float x = fl[c] / tsafe;
      ov[c] = (key_of(x) >= TKey) ? (x - Mx - lZ2) : NEG_INF;
    }
    float4 o; o.x = ov[0]; o.y = ov[1]; o.z = ov[2]; o.w = ov[3];
    O4[i] = o;
  }

  // ---------------- small per-row outputs ----------------------------------
  if (tid == 0) {
    int sampled = (temp < SAMP_EPS) ? greedyIdx : randIdx;
    ((int*)out)[row] = sampled;
    out[OUT_TLP + row] = logits[base + sampled] - lse;
#pragma unroll
    for (int j = 0; j < 8; ++j) {
      out[OUT_T8V + row * 8 + j] = s_w8v[j] - lse;
      ((int*)out)[OUT_T8I + row * 8 + j] = s_w8i[j];
    }
  }
}

extern "C" void kernel_launch(void* const* d_in, const int* in_sizes, int n_in,
                              void* d_out, int out_size, void* d_ws, size_t ws_size,
                              hipStream_t stream) {
  const float* logits      = (const float*)d_in[0];
  const float* temperature = (const float*)d_in[1];
  const int*   top_k       = (const int*)d_in[2];
  const float* top_p       = (const float*)d_in[3];
  const float* noise       = (const float*)d_in[4];
  sampler_kernel<<<BROWS, NT, 0, stream>>>(logits, temperature, top_k, top_p,
                                           noise, (float*)d_out);
  (void)in_sizes; (void)n_in; (void)out_size; (void)d_ws; (void)ws_size;
}